// RoIHeads_29506425324310
// MI455X (gfx1250) — compile-verified
//
#include <hip/hip_runtime.h>
#include <cstdint>

#define DEVFN __device__ __forceinline__

constexpr int   kN      = 50000;
constexpr int   kC      = 91;
constexpr int   KPRE    = 1000;
constexpr int   DET     = 100;
constexpr float SCORE_T = 0.05f;
constexpr float NMS_T   = 0.5f;
constexpr float MINSZ   = 0.01f;
constexpr float LOG_CLAMP = 4.135166556742356f;   // log(1000/16)
constexpr int   NBINS   = 16384;
constexpr int   SELCAP  = 4096;

// ---------------- workspace layout (bytes) ----------------
constexpr size_t OFF_HIST = 0;                               // NBINS u32
constexpr size_t OFF_CNT  = OFF_HIST + (size_t)NBINS * 4;    // 8 u32 counters (pad 64B)
constexpr size_t OFF_SEL  = OFF_CNT  + 64;                   // SELCAP uint2
constexpr size_t OFF_TOPS = OFF_SEL  + (size_t)SELCAP * 8;   // 1024 f32
constexpr size_t OFF_TOPI = OFF_TOPS + 4096;                 // 1024 u32
constexpr size_t OFF_BOX  = OFF_TOPI + 4096;                 // KPRE float4
constexpr size_t OFF_OFFB = OFF_BOX  + (size_t)KPRE * 16;    // KPRE float4 (offset boxes)
constexpr size_t OFF_LAB  = OFF_OFFB + (size_t)KPRE * 16;    // KPRE i32
constexpr size_t OFF_MASK = OFF_LAB  + (size_t)KPRE * 4;     // KPRE*32 u32 suppression bits
constexpr size_t OFF_CAND = OFF_MASK + (size_t)KPRE * 32 * 4;// candidate uint2 buffer (rest)

// ---------------- CDNA5 async global->LDS path ----------------
#if defined(__AMDGCN__) && __has_builtin(__builtin_amdgcn_global_load_async_to_lds_b128)
#define HAVE_ASYNC_B128 1
#endif
#if defined(__AMDGCN__) && __has_builtin(__builtin_amdgcn_s_wait_asynccnt)
#define HAVE_WAIT_ASYNC 1
#endif

#if defined(__AMDGCN__)
typedef int v4i __attribute__((ext_vector_type(4)));
typedef __attribute__((address_space(1))) v4i* g_v4i_p;   // global int4*
typedef __attribute__((address_space(3))) v4i* l_v4i_p;   // LDS int4*
#endif

DEVFN void lds_copy16_async(float4* dst_lds, const float4* src_glb) {
#if defined(HAVE_ASYNC_B128)
  __builtin_amdgcn_global_load_async_to_lds_b128((g_v4i_p)src_glb, (l_v4i_p)dst_lds, 0, 0);
#else
  *dst_lds = *src_glb;
#endif
}

DEVFN void lds_wait_async() {
#if defined(HAVE_WAIT_ASYNC)
  __builtin_amdgcn_s_wait_asynccnt(0);
#elif defined(HAVE_ASYNC_B128)
  asm volatile("s_wait_asynccnt 0" ::: "memory");
#endif
}

// ---------------- shared helpers ----------------
DEVFN uint32_t score_bin(float s) {
  uint32_t b = (uint32_t)(s * (float)NBINS);
  return b >= (uint32_t)NBINS ? (uint32_t)(NBINS - 1) : b;
}

// torchvision BoxCoder.decode_single (w=(10,10,5,5)) + clip_boxes_to_image
DEVFN float4 decode_clip(float4 rel, float4 p, float imw, float imh) {
  float pw = p.z - p.x, ph = p.w - p.y;
  float cx = p.x + 0.5f * pw, cy = p.y + 0.5f * ph;
  float dx = rel.x * 0.1f, dy = rel.y * 0.1f;
  float dw = fminf(rel.z * 0.2f, LOG_CLAMP);
  float dh = fminf(rel.w * 0.2f, LOG_CLAMP);
  float pcx = dx * pw + cx, pcy = dy * ph + cy;
  float ppw = __expf(dw) * pw, pph = __expf(dh) * ph;
  float x1 = fminf(fmaxf(pcx - 0.5f * ppw, 0.0f), imw);
  float y1 = fminf(fmaxf(pcy - 0.5f * pph, 0.0f), imh);
  float x2 = fminf(fmaxf(pcx + 0.5f * ppw, 0.0f), imw);
  float y2 = fminf(fmaxf(pcy + 0.5f * pph, 0.0f), imh);
  return make_float4(x1, y1, x2, y2);
}

// ---------------- K0: init ----------------
__global__ __launch_bounds__(256) void k_init(uint32_t* hist, uint32_t* counters, float* out) {
  int i = blockIdx.x * blockDim.x + threadIdx.x;
  if (i < NBINS) hist[i] = 0u;
  if (i < 8)     counters[i] = 0u;
  if (i < 600)   out[i] = 0.0f;
}

// ---------------- K1: streaming softmax + decode + threshold ----------------
// one wave32 per proposal row; single pass over the large inputs
__global__ __launch_bounds__(256) void k_score_decode(
    const float* __restrict__ logits, const float4* __restrict__ reg4,
    const float4* __restrict__ prop4,
    const int* __restrict__ imh_p, const int* __restrict__ imw_p,
    uint32_t* __restrict__ hist, uint32_t* __restrict__ counters,
    uint2* __restrict__ cand, uint32_t cand_cap) {
  const int wid  = (int)((blockIdx.x * blockDim.x + threadIdx.x) >> 5);
  const int lane = (int)(threadIdx.x & 31);
  if (wid >= kN) return;
  const float imw = (float)imw_p[0], imh = (float)imh_p[0];

  const float* row = logits + (size_t)wid * kC;
  if (lane == 0 && wid + 64 < kN)
    __builtin_prefetch((const void*)(reg4 + (size_t)(wid + 64) * kC), 0, 0);

  // classes: lane, lane+32, lane+64 (91 total)
  float x0 = row[lane];
  float x1 = row[lane + 32];
  float x2 = (lane < kC - 64) ? row[lane + 64] : -INFINITY;
  float m = fmaxf(fmaxf(x0, x1), x2);
  #pragma unroll
  for (int off = 16; off > 0; off >>= 1) m = fmaxf(m, __shfl_xor(m, off, 32));
  float e0 = __expf(x0 - m), e1 = __expf(x1 - m);
  float e2 = (lane < kC - 64) ? __expf(x2 - m) : 0.0f;
  float ssum = e0 + e1 + e2;
  #pragma unroll
  for (int off = 16; off > 0; off >>= 1) ssum += __shfl_xor(ssum, off, 32);
  const float inv = 1.0f / ssum;

  const float4 p = prop4[wid];
  const float sc[3] = {e0 * inv, e1 * inv, e2 * inv};
  #pragma unroll
  for (int k = 0; k < 3; ++k) {
    const int j = lane + 32 * k;
    const float s = sc[k];
    if (j >= 1 && j < kC && s > SCORE_T) {
      const float4 rel = reg4[(size_t)wid * kC + j];
      const float4 b = decode_clip(rel, p, imw, imh);
      const float w = b.z - b.x, h = b.w - b.y;
      if (w >= MINSZ && h >= MINSZ) {
        atomicAdd(&hist[score_bin(s)], 1u);
        uint32_t idx = atomicAdd(&counters[0], 1u);
        if (idx < cand_cap)
          cand[idx] = make_uint2(__float_as_uint(s), (uint32_t)wid * (uint32_t)kC + (uint32_t)j);
      }
    }
  }
}

// ---------------- K2: histogram scan -> top-1000 score-bin threshold ----------------
__global__ void k_scan_hist(const uint32_t* __restrict__ hist, uint32_t* __restrict__ counters) {
  if (threadIdx.x == 0 && blockIdx.x == 0) {
    uint32_t cum = 0;
    int b = NBINS - 1;
    for (; b > 0; --b) {
      cum += hist[b];
      if (cum >= (uint32_t)KPRE) break;
    }
    counters[2] = (uint32_t)b;  // select all candidates with bin >= b
  }
}

// ---------------- K3: gather candidates above threshold bin ----------------
__global__ __launch_bounds__(256) void k_gather(
    const uint2* __restrict__ cand, uint32_t* __restrict__ counters,
    uint2* __restrict__ sel, uint32_t cand_cap) {
  uint32_t total = counters[0];
  if (total > cand_cap) total = cand_cap;
  const uint32_t B = counters[2];
  const uint32_t stride = gridDim.x * blockDim.x;
  for (uint32_t i = blockIdx.x * blockDim.x + threadIdx.x; i < total; i += stride) {
    const uint2 c = cand[i];
    if (score_bin(__uint_as_float(c.x)) >= B) {
      uint32_t pos = atomicAdd(&counters[1], 1u);
      if (pos < (uint32_t)SELCAP) sel[pos] = c;
    }
  }
}

// ---------------- K4: single-block bitonic sort (descending by score) ----------------
__global__ __launch_bounds__(1024) void k_sort(
    const uint2* __restrict__ sel, const uint32_t* __restrict__ counters,
    float* __restrict__ topS, uint32_t* __restrict__ topI) {
  __shared__ uint32_t key[SELCAP];
  __shared__ uint32_t val[SELCAP];
  const int tid = (int)threadIdx.x;
  uint32_t n = counters[1];
  if (n > (uint32_t)SELCAP) n = (uint32_t)SELCAP;
  for (int i = tid; i < SELCAP; i += 1024) {
    if (i < (int)n) { uint2 c = sel[i]; key[i] = c.x; val[i] = c.y; }
    else            { key[i] = 0u;      val[i] = 0u; }
  }
  __syncthreads();
  for (int k = 2; k <= SELCAP; k <<= 1) {
    for (int j = k >> 1; j > 0; j >>= 1) {
      for (int t = tid; t < SELCAP; t += 1024) {
        const int ixj = t ^ j;
        if (ixj > t) {
          const bool desc = ((t & k) == 0);
          const uint32_t a = key[t], b = key[ixj];
          const bool sw = desc ? (a < b) : (a > b);
          if (sw) {
            key[t] = b; key[ixj] = a;
            const uint32_t tv = val[t]; val[t] = val[ixj]; val[ixj] = tv;
          }
        }
      }
      __syncthreads();
    }
  }
  for (int i = tid; i < KPRE; i += 1024) {
    topS[i] = __uint_as_float(key[i]);   // padded slots are 0.0 -> fail keep0 (s > 0)
    topI[i] = val[i];
  }
}

// ---------------- K5: re-decode the 1000 winners ----------------
__global__ __launch_bounds__(256) void k_decode_top(
    const float* __restrict__ topS, const uint32_t* __restrict__ topI,
    const float4* __restrict__ reg4, const float4* __restrict__ prop4,
    const int* __restrict__ imh_p, const int* __restrict__ imw_p,
    float4* __restrict__ boxesK, float4* __restrict__ offK, int* __restrict__ labK) {
  const int i = (int)(blockIdx.x * blockDim.x + threadIdx.x);
  if (i >= KPRE) return;
  const float s = topS[i];
  if (s <= 0.0f) {
    boxesK[i] = make_float4(0, 0, 0, 0);
    offK[i]   = make_float4(0, 0, 0, 0);
    labK[i]   = 0;
    return;
  }
  const float imw = (float)imw_p[0], imh = (float)imh_p[0];
  const uint32_t id = topI[i];
  const uint32_t r = id / (uint32_t)kC;
  const uint32_t j = id % (uint32_t)kC;
  const float4 b = decode_clip(reg4[(size_t)r * kC + j], prop4[r], imw, imh);
  boxesK[i] = b;
  const float off = (float)j * (imw + imh + 2.0f);   // batched-NMS class offset
  offK[i]   = make_float4(b.x + off, b.y + off, b.z + off, b.w + off);
  labK[i]   = (int)j;
}

// ---------------- K6a: parallel suppression-mask build ----------------
// mask[i*32+w] bit b = (j=32w+b) > i  &&  IoU(box_i, box_j) > NMS_T
// (independent of keep state: clearing an already-cleared keep bit is a no-op,
//  so this is exactly the reference's sup = keep[i] & (iou>T) & (j>i) recursion)
__global__ __launch_bounds__(256) void k_nms_mask(
    const float4* __restrict__ offK, uint32_t* __restrict__ mask) {
  __shared__ float4 ob[KPRE];
  __shared__ float  area[KPRE];
  const int tid = (int)threadIdx.x;

  // CDNA5 async global -> LDS staging of the NMS working set
  for (int i = tid; i < KPRE; i += 256) lds_copy16_async(&ob[i], &offK[i]);
  lds_wait_async();
  __syncthreads();
  for (int i = tid; i < KPRE; i += 256) {
    const float4 b = ob[i];
    area[i] = fmaxf(b.z - b.x, 0.0f) * fmaxf(b.w - b.y, 0.0f);
  }
  __syncthreads();

  const int t = (int)(blockIdx.x * 256 + tid);   // t in [0, KPRE*32)
  const int i = t >> 5, w = t & 31;
  if (i >= KPRE) return;
  const float4 a = ob[i];
  const float ai = area[i];
  uint32_t bits = 0u;
  const int j0 = w * 32;
  #pragma unroll 4
  for (int b = 0; b < 32; ++b) {
    const int j = j0 + b;
    if (j > i && j < KPRE) {
      const float4 q = ob[j];
      const float ix = fminf(a.z, q.z) - fmaxf(a.x, q.x);
      const float iy = fminf(a.w, q.w) - fmaxf(a.y, q.y);
      const float inter = fmaxf(ix, 0.0f) * fmaxf(iy, 0.0f);
      const float iou = inter / (ai + area[j] - inter + 1e-7f);
      if (iou > NMS_T) bits |= (1u << b);
    }
  }
  mask[i * 32 + w] = bits;
}

// ---------------- K6b: single-wave serial NMS reduce + compact + output ----------------
// keep bits live in 32 lanes x 32 bits; per row: 1 shfl broadcast + 1 andn.
// mask rows software-pipelined 16 deep (constant-index rotation, no scratch).
__global__ __launch_bounds__(32) void k_nms_reduce_out(
    const uint32_t* __restrict__ mask, const float* __restrict__ topS,
    const float4* __restrict__ boxesK, const int* __restrict__ labK,
    float* __restrict__ out) {
  __shared__ uint32_t kw[32];
  __shared__ int selo[DET];
  __shared__ int scnt;
  const int lane = (int)threadIdx.x;

  // lane l holds keep bits for j = 32l .. 32l+31
  uint32_t keepw = 0u;
  #pragma unroll
  for (int b = 0; b < 32; ++b) {
    const int j = lane * 32 + b;
    if (j < KPRE && topS[j] > 0.0f) keepw |= (1u << b);
  }

  constexpr int PF = 16;
  uint32_t mbuf[PF];
  #pragma unroll
  for (int d = 0; d < PF; ++d) mbuf[d] = mask[d * 32 + lane];

  for (int i0 = 0; i0 < KPRE; i0 += PF) {
    #pragma unroll
    for (int d = 0; d < PF; ++d) {
      const int i = i0 + d;
      const uint32_t cur = __shfl(keepw, i >> 5, 32);
      if (i < KPRE && ((cur >> (i & 31)) & 1u)) keepw &= ~mbuf[d];
      const int nx = i + PF;
      mbuf[d] = (nx < KPRE) ? mask[nx * 32 + lane] : 0u;
    }
  }

  kw[lane] = keepw;
  __syncthreads();
  if (lane == 0) {
    int c = 0;
    for (int j = 0; j < KPRE && c < DET; ++j)
      if ((kw[j >> 5] >> (j & 31)) & 1u) selo[c++] = j;
    scnt = c;
  }
  __syncthreads();

  // out layout: boxes[100*4] | scores[100] | labels[100]  (600 floats)
  for (int k = lane; k < DET; k += 32) {
    if (k < scnt) {
      const int i = selo[k];
      const float4 b = boxesK[i];
      out[4 * k + 0] = b.x;
      out[4 * k + 1] = b.y;
      out[4 * k + 2] = b.z;
      out[4 * k + 3] = b.w;
      out[400 + k] = topS[i];
      out[500 + k] = (float)labK[i];
    } else {
      out[4 * k + 0] = 0.0f; out[4 * k + 1] = 0.0f;
      out[4 * k + 2] = 0.0f; out[4 * k + 3] = 0.0f;
      out[400 + k] = 0.0f;
      out[500 + k] = 0.0f;
    }
  }
}

// ---------------- launch ----------------
extern "C" void kernel_launch(void* const* d_in, const int* in_sizes, int n_in,
                              void* d_out, int out_size, void* d_ws, size_t ws_size,
                              hipStream_t stream) {
  (void)in_sizes; (void)n_in; (void)out_size;
  const float*  logits = (const float*)d_in[0];
  const float4* reg4   = (const float4*)d_in[1];
  const float4* prop4  = (const float4*)d_in[2];
  const int*    imh    = (const int*)d_in[3];
  const int*    imw    = (const int*)d_in[4];
  float*        out    = (float*)d_out;

  char* ws = (char*)d_ws;
  uint32_t* hist     = (uint32_t*)(ws + OFF_HIST);
  uint32_t* counters = (uint32_t*)(ws + OFF_CNT);
  uint2*    sel      = (uint2*)   (ws + OFF_SEL);
  float*    topS     = (float*)   (ws + OFF_TOPS);
  uint32_t* topI     = (uint32_t*)(ws + OFF_TOPI);
  float4*   boxesK   = (float4*)  (ws + OFF_BOX);
  float4*   offK     = (float4*)  (ws + OFF_OFFB);
  int*      labK     = (int*)     (ws + OFF_LAB);
  uint32_t* maskp    = (uint32_t*)(ws + OFF_MASK);
  uint2*    cand     = (uint2*)   (ws + OFF_CAND);

  size_t cand_cap = 0;
  if (ws_size > OFF_CAND + sizeof(uint2))
    cand_cap = (ws_size - OFF_CAND) / sizeof(uint2);
  if (cand_cap > (size_t)(8u << 20)) cand_cap = (size_t)(8u << 20);

  k_init<<<(NBINS + 255) / 256, 256, 0, stream>>>(hist, counters, out);
  k_score_decode<<<(kN * 32 + 255) / 256, 256, 0, stream>>>(
      logits, reg4, prop4, imh, imw, hist, counters, cand, (uint32_t)cand_cap);
  k_scan_hist<<<1, 64, 0, stream>>>(hist, counters);
  k_gather<<<512, 256, 0, stream>>>(cand, counters, sel, (uint32_t)cand_cap);
  k_sort<<<1, 1024, 0, stream>>>(sel, counters, topS, topI);
  k_decode_top<<<(KPRE + 255) / 256, 256, 0, stream>>>(
      topS, topI, reg4, prop4, imh, imw, boxesK, offK, labK);
  k_nms_mask<<<(KPRE * 32 + 255) / 256, 256, 0, stream>>>(offK, maskp);
  k_nms_reduce_out<<<1, 32, 0, stream>>>(maskp, topS, boxesK, labK, out);
}